// FuncSelfAttention_67035849556506
// MI455X (gfx1250) — compile-verified
//
#include <hip/hip_runtime.h>
#include <math.h>

// Problem constants (fixed by reference setup_inputs)
#define BDIM 8
#define SDIM 64
#define CDIM 64
#define HDIM 32
#define WDIM 32
#define NH 8
#define HD 8
#define T7 7
#define K1 (CDIM*T7)          // 448  (coeff columns per field set)
#define N1 (3*CDIM*T7)        // 1344 (qkv output columns)
#define BS (BDIM*SDIM)        // 512
#define PLANE (HDIM*WDIM)     // 1024

typedef float v2f __attribute__((ext_vector_type(2)));
typedef float v8f __attribute__((ext_vector_type(8)));

// ---------------------------------------------------------------------------
// Kernel 1: analysis — project each (b,s,c) 32x32 plane onto the 4 rfft2 modes
// (kx,ky in {0,1}^2), emitting 7 reals: ReA, ReB, ImB, ReC, ImC, ReD, ImD.
// One wave per plane.  Lanes map to LINEAR float4 chunks so every wave load is
// 512 contiguous bytes (4 cachelines) -> full-bandwidth pattern for 23.3 TB/s.
// chunk c = it*32+lane covers h = c>>3, w = (c&7)*4 .. +3 (w phase is
// lane-invariant across iterations -> cos/sin of w preloaded in registers).
// ---------------------------------------------------------------------------
__global__ void analysis_kernel(const float* __restrict__ seq, float* __restrict__ Xc) {
  __shared__ float CW[32], SW[32];
  int tid = threadIdx.x;
  if (tid < 32) {
    float ang = 6.28318530717958647692f * (float)tid / 32.0f;
    float s, c;
    __sincosf(ang, &s, &c);
    SW[tid] = s; CW[tid] = c;
  }
  __syncthreads();
  int wave = tid >> 5;
  int lane = tid & 31;
  int p = blockIdx.x * 8 + wave;                // plane over BS*C = 32768
  const float4* src = reinterpret_cast<const float4*>(seq + (size_t)p * PLANE);
  int wb = (lane & 7) * 4;
  float cw0 = CW[wb],   cw1 = CW[wb+1], cw2 = CW[wb+2], cw3 = CW[wb+3];
  float sw0 = SW[wb],   sw1 = SW[wb+1], sw2 = SW[wb+2], sw3 = SW[wb+3];
  float r0 = 0.f, r1 = 0.f, r2 = 0.f, r3 = 0.f, r4 = 0.f, r5 = 0.f, r6 = 0.f;
  #pragma unroll
  for (int it = 0; it < 8; ++it) {
    float4 x = src[it * 32 + lane];
    int h = it * 4 + (lane >> 3);
    float ch = CW[h], sh = SW[h];
    float s0 = x.x + x.y + x.z + x.w;
    float sc = x.x*cw0 + x.y*cw1 + x.z*cw2 + x.w*cw3;
    float ss = x.x*sw0 + x.y*sw1 + x.z*sw2 + x.w*sw3;
    // F[kx,ky] = sum x * e^{-i(kx*th + ky*phw)}
    r0 += s0;                    // ReA
    r1 += s0 * ch;               // ReB
    r2 -= s0 * sh;               // ImB
    r3 += sc;                    // ReC
    r4 -= ss;                    // ImC
    r5 += sc * ch - ss * sh;     // ReD
    r6 -= sc * sh + ss * ch;     // ImD
  }
  #pragma unroll
  for (int off = 16; off > 0; off >>= 1) {
    r0 += __shfl_xor(r0, off, 32);
    r1 += __shfl_xor(r1, off, 32);
    r2 += __shfl_xor(r2, off, 32);
    r3 += __shfl_xor(r3, off, 32);
    r4 += __shfl_xor(r4, off, 32);
    r5 += __shfl_xor(r5, off, 32);
    r6 += __shfl_xor(r6, off, 32);
  }
  if (lane == 0) {
    float* dst = Xc + (size_t)(p >> 6) * K1 + (p & 63) * T7;  // row=(b,s), col=c*7+t
    dst[0]=r0; dst[1]=r1; dst[2]=r2; dst[3]=r3; dst[4]=r4; dst[5]=r5; dst[6]=r6;
  }
}

// ---------------------------------------------------------------------------
// Kernel 2: build combined real weight matrix from complex per-mode weights.
// Per (c,o) a 7x7 block (block-diag 1+2+2+2) encoding the complex multiply.
// bscale = 0.5 for the second conv (rfft2 round-trip halves the F10 mode).
// ---------------------------------------------------------------------------
__global__ void build_mat_kernel(const float* __restrict__ wr, const float* __restrict__ wi,
                                 float* __restrict__ M, int O, float bscale) {
  int t = blockIdx.x * blockDim.x + threadIdx.x;
  if (t >= CDIM * O) return;
  int c = t / O, o = t % O;
  const float* pr = wr + (size_t)(c * O + o) * 4;   // [kx][ky] row-major
  const float* pi = wi + (size_t)(c * O + o) * 4;
  float wr00 = pr[0], wr01 = pr[1], wr10 = pr[2], wr11 = pr[3];
  float wi01 = pi[1], wi10 = pi[2], wi11 = pi[3];
  int ld = O * T7;
  float* base = M + (size_t)(c * T7) * ld + o * T7;
  #pragma unroll
  for (int a = 0; a < 7; ++a)
    #pragma unroll
    for (int b = 0; b < 7; ++b)
      base[(size_t)a * ld + b] = 0.f;
  base[0*(size_t)ld+0] =  wr00;                         // ReA' = ReA*wr00
  base[1*(size_t)ld+1] =  wr10 * bscale;                // ReB' = ReB*wr - ImB*wi
  base[1*(size_t)ld+2] =  wi10 * bscale;                // ImB' = ReB*wi + ImB*wr
  base[2*(size_t)ld+1] = -wi10 * bscale;
  base[2*(size_t)ld+2] =  wr10 * bscale;
  base[3*(size_t)ld+3] =  wr01;
  base[3*(size_t)ld+4] =  wi01;
  base[4*(size_t)ld+3] = -wi01;
  base[4*(size_t)ld+4] =  wr01;
  base[5*(size_t)ld+5] =  wr11;
  base[5*(size_t)ld+6] =  wi11;
  base[6*(size_t)ld+5] = -wi11;
  base[6*(size_t)ld+6] =  wr11;
}

// ---------------------------------------------------------------------------
// Kernel 3: f32 GEMM with V_WMMA_F32_16X16X4_F32.  D = A[MxK] * B[KxN].
// One 16x16 output tile per wave; A/B operands loaded directly in the ISA
// lane layout (M/N = lane&15, K split by lane half, 2 f32 per lane).
// ---------------------------------------------------------------------------
__global__ void gemm_f32_wmma(const float* __restrict__ A, const float* __restrict__ Bm,
                              float* __restrict__ D, int M, int N, int K) {
  int wave = (blockIdx.x * blockDim.x + threadIdx.x) >> 5;
  int lane = threadIdx.x & 31;
  int ntiles = N >> 4;
  int tm = (wave / ntiles) << 4;
  int tn = (wave % ntiles) << 4;
  if (tm >= M) return;
  int half = lane >> 4;       // selects K pair {0,1} vs {2,3}
  int l15 = lane & 15;
  v8f c = {0.f,0.f,0.f,0.f,0.f,0.f,0.f,0.f};
  const float* arow = A + (size_t)(tm + l15) * K + half * 2;
  const float* bcol = Bm + (size_t)(half * 2) * N + tn + l15;
  for (int kb = 0; kb < K; kb += 4) {
    v2f a, b;
    a.x = arow[kb];
    a.y = arow[kb + 1];
    b.x = bcol[(size_t)kb * N];
    b.y = bcol[(size_t)kb * N + N];
    c = __builtin_amdgcn_wmma_f32_16x16x4_f32(false, a, false, b, (short)0, c, false, false);
  }
  #pragma unroll
  for (int r = 0; r < 8; ++r)
    D[(size_t)(tm + r + half * 8) * N + tn + l15] = c[r];
}

// ---------------------------------------------------------------------------
// Kernel 4: Swin-v2 style continuous position bias MLP -> bias[nh=8,64,64].
// ---------------------------------------------------------------------------
__global__ void cpb_bias_kernel(const float* __restrict__ w1, const float* __restrict__ b1,
                                const float* __restrict__ w2, float* __restrict__ bias) {
  int i = blockIdx.x, j = threadIdx.x;
  float r0 = (float)(i >> 3) - (float)(j >> 3);
  float r1 = (float)(i & 7) - (float)(j & 7);
  r0 = copysignf(log2f(1.f + fabsf(r0)), r0);
  r1 = copysignf(log2f(1.f + fabsf(r1)), r1);
  float acc[NH];
  #pragma unroll
  for (int k = 0; k < NH; ++k) acc[k] = 0.f;
  for (int e = 0; e < 64; ++e) {
    float h = fmaxf(r0 * w1[e] + r1 * w1[64 + e] + b1[e], 0.f);
    #pragma unroll
    for (int k = 0; k < NH; ++k) acc[k] += h * w2[e * NH + k];
  }
  #pragma unroll
  for (int k = 0; k < NH; ++k) bias[k * 4096 + i * 64 + j] = acc[k];
}

// ---------------------------------------------------------------------------
// Kernel 5: attention in 56-dim coefficient space per (b,head), 256 threads.
// Quadrature inner product folded into Q via g-weights {1,.5,.5,2,2,2,2}*alpha.
// Phase A: all 64x64 scores -> LDS (each thread: 16 scores).
// Phase B: row softmax (64 threads).
// Phase C: output GEMV (each thread: 14 of the 64x56 outputs).
// ---------------------------------------------------------------------------
__global__ void attention_kernel(const float* __restrict__ qkvm,
                                 const float* __restrict__ bias,
                                 float* __restrict__ Uout) {
  __shared__ float Qs[64][57];
  __shared__ float Ks[64][57];
  __shared__ float Vs[64][57];
  __shared__ float Ps[64][65];
  __shared__ float RS[64];
  int b = blockIdx.x >> 3;
  int head = blockIdx.x & 7;
  int tid = threadIdx.x;          // 0..255
  int i = tid >> 2;               // 0..63 (row owned by this thread)
  int qtr = tid & 3;              // 0..3
  const float alpha = 9.5367431640625e-07f * 0.35355339059327373f; // 1/(HW)^2/sqrt(hd)
  size_t rowbase = (size_t)(b * 64 + i) * N1 + (size_t)head * 56;
  // cooperative load: each thread loads 14 coefficients of q,k,v for row i
  for (int t = qtr * 14; t < qtr * 14 + 14; ++t) {
    int tt = t - (t / 7) * 7;
    float gw = (tt == 0) ? 1.f : ((tt < 3) ? 0.5f : 2.f);
    Qs[i][t] = qkvm[rowbase + t] * (gw * alpha);
    Ks[i][t] = qkvm[rowbase + 448 + t];
    Vs[i][t] = qkvm[rowbase + 896 + t];
  }
  __syncthreads();
  // scores
  const float* bsr = bias + head * 4096 + i * 64;
  for (int jj = 0; jj < 16; ++jj) {
    int j = qtr * 16 + jj;
    float s = bsr[j];
    #pragma unroll
    for (int t = 0; t < 56; ++t) s += Qs[i][t] * Ks[j][t];
    Ps[i][j] = s;
  }
  __syncthreads();
  // softmax per row
  if (tid < 64) {
    float m = -1e30f;
    #pragma unroll
    for (int j = 0; j < 64; ++j) m = fmaxf(m, Ps[tid][j]);
    float sum = 0.f;
    #pragma unroll
    for (int j = 0; j < 64; ++j) { float p = __expf(Ps[tid][j] - m); Ps[tid][j] = p; sum += p; }
    RS[tid] = 1.f / sum;
  }
  __syncthreads();
  // output: out[i][t] = (1/sum_i) * sum_j P[i][j] * V[j][t]
  float* orow = Uout + (size_t)(b * 64 + i) * K1 + (size_t)head * 56;
  float rs = RS[i];
  for (int t = qtr * 14; t < qtr * 14 + 14; ++t) {
    float acc = 0.f;
    #pragma unroll
    for (int j = 0; j < 64; ++j) acc += Ps[i][j] * Vs[j][t];
    orow[t] = acc * rs;
  }
}

// ---------------------------------------------------------------------------
// Kernel 6: synthesis — expand 7 coefficients back to the 32x32 plane:
// y = (1/HW)[ReA + ReB cosθ - ImB sinθ + 2(ReC cosφ - ImC sinφ)
//            + 2(ReD cos(θ+φ) - ImD sin(θ+φ))]
// Same linear-chunk lane mapping as analysis -> 512B contiguous wave stores.
// ---------------------------------------------------------------------------
__global__ void synthesis_kernel(const float* __restrict__ O2, float* __restrict__ y) {
  __shared__ float CW[32], SW[32];
  int tid = threadIdx.x;
  if (tid < 32) {
    float ang = 6.28318530717958647692f * (float)tid / 32.0f;
    float s, c;
    __sincosf(ang, &s, &c);
    SW[tid] = s; CW[tid] = c;
  }
  __syncthreads();
  int wave = tid >> 5, lane = tid & 31;
  int p = blockIdx.x * 8 + wave;
  const float* u = O2 + (size_t)(p >> 6) * K1 + (p & 63) * T7;
  float u0=u[0], u1=u[1], u2=u[2], u3=u[3], u4=u[4], u5=u[5], u6=u[6];
  const float inv = 1.0f / 1024.0f;
  int wb = (lane & 7) * 4;
  float cw0 = CW[wb],   cw1 = CW[wb+1], cw2 = CW[wb+2], cw3 = CW[wb+3];
  float sw0 = SW[wb],   sw1 = SW[wb+1], sw2 = SW[wb+2], sw3 = SW[wb+3];
  float4* dst = reinterpret_cast<float4*>(y + (size_t)p * PLANE);
  #pragma unroll
  for (int it = 0; it < 8; ++it) {
    int h = it * 4 + (lane >> 3);
    float ch = CW[h], sh = SW[h];
    float p0 = (u0 + u1*ch - u2*sh) * inv;
    float qc = 2.f * ( u3 + u5*ch - u6*sh) * inv;
    float qs = 2.f * (-u4 - u5*sh - u6*ch) * inv;
    float4 o;
    o.x = p0 + qc*cw0 + qs*sw0;
    o.y = p0 + qc*cw1 + qs*sw1;
    o.z = p0 + qc*cw2 + qs*sw2;
    o.w = p0 + qc*cw3 + qs*sw3;
    dst[it * 32 + lane] = o;
  }
}

// ---------------------------------------------------------------------------
extern "C" void kernel_launch(void* const* d_in, const int* in_sizes, int n_in,
                              void* d_out, int out_size, void* d_ws, size_t ws_size,
                              hipStream_t stream) {
  (void)in_sizes; (void)n_in; (void)out_size; (void)ws_size;
  const float* seq    = (const float*)d_in[0];
  const float* wqr    = (const float*)d_in[1];
  const float* wqi    = (const float*)d_in[2];
  const float* wor    = (const float*)d_in[3];
  const float* woi    = (const float*)d_in[4];
  const float* cpb_w1 = (const float*)d_in[5];
  const float* cpb_b1 = (const float*)d_in[6];
  const float* cpb_w2 = (const float*)d_in[7];

  char* ws = (char*)d_ws;
  float* Xc   = (float*)(ws + 0);        // [512,448]   917504 B
  float* M1   = (float*)(ws + 917504);   // [448,1344] 2408448 B
  float* QKV  = (float*)(ws + 3325952);  // [512,1344] 2752512 B
  float* BIAS = (float*)(ws + 6078464);  // [8,64,64]   131072 B
  float* U    = (float*)(ws + 6209536);  // [512,448]   917504 B
  float* M2   = (float*)(ws + 7127040);  // [448,448]   802816 B
  float* O2   = (float*)(ws + 7929856);  // [512,448]   917504 B

  analysis_kernel<<<4096, 256, 0, stream>>>(seq, Xc);
  build_mat_kernel<<<48, 256, 0, stream>>>(wqr, wqi, M1, 3 * CDIM, 1.0f);
  build_mat_kernel<<<16, 256, 0, stream>>>(wor, woi, M2, CDIM, 0.5f);
  cpb_bias_kernel<<<64, 64, 0, stream>>>(cpb_w1, cpb_b1, cpb_w2, BIAS);
  // QKV spectral conv as real GEMM: [512x448] x [448x1344]
  gemm_f32_wmma<<<336, 256, 0, stream>>>(Xc, M1, QKV, BS, N1, K1);
  attention_kernel<<<64, 256, 0, stream>>>(QKV, BIAS, U);
  // output spectral conv: [512x448] x [448x448]
  gemm_f32_wmma<<<112, 256, 0, stream>>>(U, M2, O2, BS, K1, K1);
  synthesis_kernel<<<4096, 256, 0, stream>>>(O2, (float*)d_out);
}